// SRR_CG_45509473468736
// MI455X (gfx1250) — compile-verified
//
#include <hip/hip_runtime.h>

// SRR-CG for MI455X (gfx1250, wave32).
// - Fused AtA: gather(45 taps x 8 corners) then splat via global_atomic_add_f32 (L2-resolved).
// - Rotated base grid precomputed ONCE with v_wmma_f32_16x16x4_f32 (true skinny GEMM:
//   16384x4 base points x packed 3x3 rotations), reused by all 23 operator passes.
// - All CG scalars on-device; fixed kernel sequence -> graph-capture safe.

#define DIMV 128
#define NVOX (128*128*128)          // volume voxels
#define NSL  48                     // slices
#define SPIX (128*128)              // pixels per slice
#define NPIX (NSL*SPIX)             // total slice pixels
#define NTAP 45                     // 5x3x3 PSF taps

typedef __attribute__((ext_vector_type(2))) float v2f;
typedef __attribute__((ext_vector_type(8))) float v8f;

// ---------------- tiny vector kernels ----------------
__global__ void k_zero(float* __restrict__ b, int n){
  for (int i = blockIdx.x*blockDim.x + threadIdx.x; i < n; i += gridDim.x*blockDim.x) b[i] = 0.f;
}
__global__ void k_copy(float* __restrict__ d, const float* __restrict__ s, int n){
  for (int i = blockIdx.x*blockDim.x + threadIdx.x; i < n; i += gridDim.x*blockDim.x) d[i] = s[i];
}
__global__ void k_relu(float* __restrict__ x, int n){
  for (int i = blockIdx.x*blockDim.x + threadIdx.x; i < n; i += gridDim.x*blockDim.x) x[i] = fmaxf(x[i], 0.f);
}

// scal layout: [0]=rr [1]=pAp [2]=rr_new [3]=alpha [4]=beta
__global__ void k_alpha(float* s){ s[3] = s[0] / s[1]; }
__global__ void k_beta (float* s){ s[4] = s[2] / s[0]; s[0] = s[2]; }

// x += a*p ; optionally r -= a*Ap
__global__ void k_xr(float* __restrict__ x, float* __restrict__ r,
                     const float* __restrict__ p, const float* __restrict__ Ap,
                     const float* __restrict__ scal, int upd_r, int n){
  float a = scal[3];
  for (int i = blockIdx.x*blockDim.x + threadIdx.x; i < n; i += gridDim.x*blockDim.x){
    x[i] += a * p[i];
    if (upd_r) r[i] -= a * Ap[i];
  }
}
// p = r + beta*p
__global__ void k_pup(float* __restrict__ p, const float* __restrict__ r,
                      const float* __restrict__ scal, int n){
  float b = scal[4];
  for (int i = blockIdx.x*blockDim.x + threadIdx.x; i < n; i += gridDim.x*blockDim.x)
    p[i] = r[i] + b * p[i];
}

// dot(a,b) -> atomicAdd into *out (out must be pre-zeroed)
__global__ __launch_bounds__(256) void k_dot(const float* __restrict__ a, const float* __restrict__ b,
                                             float* __restrict__ out, int n){
  __shared__ float sm[256];
  float s = 0.f;
  for (int i = blockIdx.x*blockDim.x + threadIdx.x; i < n; i += gridDim.x*blockDim.x)
    s += a[i] * b[i];
  sm[threadIdx.x] = s; __syncthreads();
  for (int off = 128; off > 0; off >>= 1){
    if ((int)threadIdx.x < off) sm[threadIdx.x] += sm[threadIdx.x + off];
    __syncthreads();
  }
  if (threadIdx.x == 0) atomicAdd(out, sm[0]);
}

// ---------------- per-(slice,tap) constants: delta = R*off + t + center, w = psf[k] ----------------
__global__ void k_delta(const float* __restrict__ trans, const float* __restrict__ psf,
                        float4* __restrict__ delta){
  int t = blockIdx.x*blockDim.x + threadIdx.x;
  if (t >= NSL*NTAP) return;
  int n = t / NTAP, k = t % NTAP;
  int kz = k / 9, ky = (k % 9) / 3, kx = k % 3;
  float ox = (float)kx - 1.f, oy = (float)ky - 1.f, oz = (float)kz - 2.f;
  const float* T = trans + n*12;
  float dx = T[0]*ox + T[1]*oy + T[2]*oz  + T[3]  + 63.5f;
  float dy = T[4]*ox + T[5]*oy + T[6]*oz  + T[7]  + 63.5f;
  float dz = T[8]*ox + T[9]*oy + T[10]*oz + T[11] + 63.5f;
  delta[t] = make_float4(dx, dy, dz, psf[k]);
}

// ---------------- rotated base grid via WMMA f32 16x16x4 ----------------
// D(16x16) = A(16x4) x B(4x16).  A rows = 16 base-grid points (x, y, 0, 0).
// B packs 5 slices: col j (<15) = column of R for slice n0+j/3, coord c=j%3:
//   B[0][j] = R[c][0], B[1][j] = R[c][1], rows 2,3 = 0 (base z == 0).
// Output layout (C/D f32): lane j<16 VGPR v = D[M=v][N=j]; lanes 16..31 hold M=v+8.
// We store each lane's 8 accumulator VGPRs contiguously into rbase[n][c][pt].
__global__ __launch_bounds__(256) void k_rbase(const float* __restrict__ trans,
                                               float* __restrict__ rbase){
  int lane  = threadIdx.x & 31;
  int gwave = blockIdx.x * 8 + (threadIdx.x >> 5);   // 10240 waves total
  int tile  = gwave & 1023;                          // 1024 tiles of 16 points
  int grp   = gwave >> 10;                           // 10 groups of <=5 slices
  int n0    = grp * 5;
  int j     = lane & 15;
  bool lo   = lane < 16;

  // A operand: lanes 0..15 carry K=0 (x), K=1 (y); lanes 16..31 carry K=2,3 = 0
  int   mg  = tile * 16 + j;                         // point index 0..16383
  float bx  = ((float)(mg & 127) - 63.5f) * 1.25f;   // base x (RATIO=1.25)
  float by  = ((float)(mg >> 7)  - 63.5f) * 1.25f;   // base y
  v2f A; A.x = lo ? bx : 0.f; A.y = lo ? by : 0.f;

  // B operand: lane holds B[lo?0:1][j] in VGPR0, B[lo?2:3][j]=0 in VGPR1
  int n = n0 + j / 3;
  int c = j - (j / 3) * 3;
  float bval = 0.f;
  if (j < 15 && n < NSL) bval = trans[n*12 + c*4 + (lo ? 0 : 1)];
  v2f B; B.x = bval; B.y = 0.f;

  v8f C = {};
#if defined(__AMDGCN__)
  v8f D = __builtin_amdgcn_wmma_f32_16x16x4_f32(false, A, false, B, (short)0, C, false, false);
#else
  v8f D = C;
#endif
  if (j < 15 && n < NSL){
    float* out = rbase + ((size_t)(n*3 + c) << 14) + (tile << 4) + (lo ? 0 : 8);
    *(v8f*)out = D;   // 32B-aligned: tile*16 floats is 64B-aligned, +8 floats is 32B-aligned
  }
}

// ---------------- trilinear gather / splat ----------------
__device__ __forceinline__ float trigather(const float* __restrict__ v,
                                           float x, float y, float z){
  float xf = floorf(x), yf = floorf(y), zf = floorf(z);
  int x0 = (int)xf, y0 = (int)yf, z0 = (int)zf;
  float fx = x - xf, fy = y - yf, fz = z - zf;
  float wx[2] = {1.f - fx, fx}, wy[2] = {1.f - fy, fy}, wz[2] = {1.f - fz, fz};
  float acc = 0.f;
  #pragma unroll
  for (int dz = 0; dz < 2; ++dz){
    int zi = z0 + dz; if ((unsigned)zi >= 128u) continue;
    #pragma unroll
    for (int dy = 0; dy < 2; ++dy){
      int yi = y0 + dy; if ((unsigned)yi >= 128u) continue;
      int base = (zi * 128 + yi) * 128;
      float wzy = wz[dz] * wy[dy];
      #pragma unroll
      for (int dx = 0; dx < 2; ++dx){
        int xi = x0 + dx; if ((unsigned)xi >= 128u) continue;
        acc += wzy * wx[dx] * v[base + xi];
      }
    }
  }
  return acc;
}

__device__ __forceinline__ void trisplat(float* __restrict__ v,
                                         float x, float y, float z, float val){
  float xf = floorf(x), yf = floorf(y), zf = floorf(z);
  int x0 = (int)xf, y0 = (int)yf, z0 = (int)zf;
  float fx = x - xf, fy = y - yf, fz = z - zf;
  float wx[2] = {1.f - fx, fx}, wy[2] = {1.f - fy, fy}, wz[2] = {1.f - fz, fz};
  #pragma unroll
  for (int dz = 0; dz < 2; ++dz){
    int zi = z0 + dz; if ((unsigned)zi >= 128u) continue;
    #pragma unroll
    for (int dy = 0; dy < 2; ++dy){
      int yi = y0 + dy; if ((unsigned)yi >= 128u) continue;
      int base = (zi * 128 + yi) * 128;
      float wzy = wz[dz] * wy[dy];
      #pragma unroll
      for (int dx = 0; dx < 2; ++dx){
        int xi = x0 + dx; if ((unsigned)xi >= 128u) continue;
        atomicAdd(&v[base + xi], val * wzy * wx[dx]);
      }
    }
  }
}

// ---------------- fused operator kernel ----------------
// srcvol != null : dst += scale * At(A(srcvol))   (fused gather+splat, per slice pixel)
// srcvol == null : dst += scale * At(srcsl)       (splat slice data; used for b = At(y))
__global__ __launch_bounds__(256) void k_op(const float* __restrict__ srcvol,
                                            const float* __restrict__ srcsl,
                                            float* __restrict__ dst,
                                            const float4* __restrict__ delta,
                                            const float* __restrict__ rbase,
                                            float scale){
  __shared__ float4 dsm[NTAP];
  int pix = blockIdx.x * 256 + threadIdx.x;      // 3072 blocks x 256 == NPIX exactly
  int n = pix >> 14;                             // block spans one slice (16384 % 256 == 0)
  int m = pix & 16383;
  if (threadIdx.x < NTAP) dsm[threadIdx.x] = delta[n * NTAP + threadIdx.x];
  __syncthreads();

  float bx = rbase[((size_t)(n*3 + 0) << 14) + m];
  float by = rbase[((size_t)(n*3 + 1) << 14) + m];
  float bz = rbase[((size_t)(n*3 + 2) << 14) + m];

  float s;
  if (srcvol){
    s = 0.f;
    for (int k = 0; k < NTAP; ++k){
      float4 d = dsm[k];
      s += d.w * trigather(srcvol, bx + d.x, by + d.y, bz + d.z);
    }
  } else {
    s = srcsl[pix];
  }
  s *= scale;
  for (int k = 0; k < NTAP; ++k){
    float4 d = dsm[k];
    trisplat(dst, bx + d.x, by + d.y, bz + d.z, s * d.w);
  }
}

// ---------------- host driver ----------------
extern "C" void kernel_launch(void* const* d_in, const int* in_sizes, int n_in,
                              void* d_out, int out_size, void* d_ws, size_t ws_size,
                              hipStream_t stream){
  const float* vol    = (const float*)d_in[0];   // 128^3
  const float* slices = (const float*)d_in[1];   // 48*128*128
  const float* trans  = (const float*)d_in[2];   // 48*3*4
  const float* psf    = (const float*)d_in[3];   // 45
  float* x = (float*)d_out;                      // CG iterate lives in d_out

  // workspace layout (floats):
  //   delta  : 48*45 float4        = 8640 floats
  //   scal   : 8
  //   rbase  : 48*3*16384          = 2359296
  //   r,p,Ap : 3 * 2097152
  // total ~34.6 MB
  float*  ws    = (float*)d_ws;
  float4* delta = (float4*)d_ws;
  float*  scal  = ws + NSL*NTAP*4;
  float*  rbase = ws + NSL*NTAP*4 + 64;          // 16/32B aligned
  float*  r     = rbase + (size_t)NSL*3*SPIX;
  float*  p     = r + NVOX;
  float*  Ap    = p + NVOX;

  const int VB = NVOX / 256;    // 8192
  const int PB = NPIX / 256;    // 3072

  // precompute (transforms fixed across all CG iterations)
  k_delta<<<(NSL*NTAP + 255)/256, 256, 0, stream>>>(trans, psf, delta);
  k_rbase<<<1280, 256, 0, stream>>>(trans, rbase);   // v_wmma_f32_16x16x4_f32

  // x = vol ; r = At(y) - AtA(x)
  k_copy<<<VB, 256, 0, stream>>>(x, vol, NVOX);
  k_zero<<<VB, 256, 0, stream>>>(r, NVOX);
  k_op<<<PB, 256, 0, stream>>>(nullptr, slices, r, delta, rbase,  1.0f);
  k_op<<<PB, 256, 0, stream>>>(x,       nullptr, r, delta, rbase, -1.0f);
  k_copy<<<VB, 256, 0, stream>>>(p, r, NVOX);
  k_zero<<<1, 64, 0, stream>>>(scal, 8);
  k_dot<<<1024, 256, 0, stream>>>(r, r, &scal[0], NVOX);   // rr

  for (int i = 0; i < 10; ++i){
    k_zero<<<VB, 256, 0, stream>>>(Ap, NVOX);
    k_op<<<PB, 256, 0, stream>>>(p, nullptr, Ap, delta, rbase, 1.0f);
    k_zero<<<1, 64, 0, stream>>>(&scal[1], 1);
    k_dot<<<1024, 256, 0, stream>>>(p, Ap, &scal[1], NVOX); // pAp
    k_alpha<<<1, 1, 0, stream>>>(scal);
    if (i == 9){
      k_xr<<<VB, 256, 0, stream>>>(x, r, p, Ap, scal, 0, NVOX);
      break;
    }
    k_xr<<<VB, 256, 0, stream>>>(x, r, p, Ap, scal, 1, NVOX);
    k_zero<<<1, 64, 0, stream>>>(&scal[2], 1);
    k_dot<<<1024, 256, 0, stream>>>(r, r, &scal[2], NVOX);  // rr_new
    k_beta<<<1, 1, 0, stream>>>(scal);
    k_pup<<<VB, 256, 0, stream>>>(p, r, scal, NVOX);
  }
  k_relu<<<VB, 256, 0, stream>>>(x, NVOX);

  (void)in_sizes; (void)n_in; (void)out_size; (void)ws_size;
}